// WarpDTI_19112604467638
// MI455X (gfx1250) — compile-verified
//
#include <hip/hip_runtime.h>
#include <cstdint>

// Problem constants (fixed by the reference: dti (2,6,64,64,64), ddf (2,3,64,64,64))
#define HD   64
#define S3   (HD*HD*HD)        // 262144 voxel stride per channel
// Tile: 4 x 8 x 8 voxels = 256 threads (8 wave32 waves / block)
#define TX 4
#define TY 8
#define TZ 8
#define HX (TX+2)              // 6
#define HY (TY+2)              // 10
#define HZ (TZ+2)              // 10
#define HALO_N   (HX*HY*HZ)    // 600 elements per ddf channel
#define HALO_TOT (3*HALO_N)    // 1800 floats = 7.2 KB LDS

// 4-byte-aligned float2 so clang emits a dword-aligned global_load_b64
typedef float f2v __attribute__((ext_vector_type(2)));
typedef f2v uf2 __attribute__((aligned(4)));

__device__ __forceinline__ int iclamp(int v, int lo, int hi) {
    return v < lo ? lo : (v > hi ? hi : v);
}

__global__ __launch_bounds__(256) void warp_dti_polar_kernel(
    const float* __restrict__ dti,
    const float* __restrict__ ddf,
    float* __restrict__ out)
{
    __shared__ float sddf[HALO_TOT];

    const int blk  = blockIdx.x;
    const int b    = blk >> 10;            // 1024 tiles per batch
    const int tile = blk & 1023;
    const int x0   = (tile >> 6) * TX;     // 16 tiles in x
    const int y0   = ((tile >> 3) & 7) * TY;
    const int z0   = (tile & 7) * TZ;

    const int tid = threadIdx.x;

    // ---------------------------------------------------------------
    // Stage ddf tile + clamped halo into LDS with gfx1250 async loads.
    // Per-lane: copies 4 bytes from the global address in gaddr to the LDS offset in ldsoff; tracked by ASYNCcnt.
    // ---------------------------------------------------------------
    const float* ddfb = ddf + (size_t)b * 3 * S3;
    for (int i = tid; i < HALO_TOT; i += 256) {
        int c  = i / HALO_N;
        int r  = i - c * HALO_N;
        int hx = r / (HY * HZ);
        int r2 = r - hx * (HY * HZ);
        int hy = r2 / HZ;
        int hz = r2 - hy * HZ;
        int gx = iclamp(x0 + hx - 1, 0, HD - 1);
        int gy = iclamp(y0 + hy - 1, 0, HD - 1);
        int gz = iclamp(z0 + hz - 1, 0, HD - 1);
        const float* gp = ddfb + ((size_t)c * S3 + ((size_t)gx * HD + gy) * HD + gz);
        uint32_t ldsoff = (uint32_t)(uintptr_t)(&sddf[i]); // AS3 offset lives in low 32 bits
        uint64_t gaddr  = (uint64_t)(uintptr_t)gp;
        asm volatile("global_load_async_to_lds_b32 %0, %1, off"
                     :: "v"(ldsoff), "v"(gaddr) : "memory");
    }
    asm volatile("s_wait_asynccnt 0" ::: "memory");
    __syncthreads();

    // ---------------------------------------------------------------
    // Per-thread voxel
    // ---------------------------------------------------------------
    const int lz = tid & 7;
    const int ly = (tid >> 3) & 7;
    const int lx = tid >> 6;               // 0..3
    const int gx = x0 + lx, gy = y0 + ly, gz = z0 + lz;

#define SD(c, x, y, z) sddf[(c)*HALO_N + ((x)*HY + (y))*HZ + (z)]

    const float u0 = SD(0, lx + 1, ly + 1, lz + 1);
    const float u1 = SD(1, lx + 1, ly + 1, lz + 1);
    const float u2 = SD(2, lx + 1, ly + 1, lz + 1);

    // Jacobian J = I + du/dx. Clamped halo + per-axis scale reproduces
    // jnp.gradient exactly: interior (f[+1]-f[-1])/2, boundary one-sided.
    const float sx = (gx == 0 || gx == HD - 1) ? 1.0f : 0.5f;
    const float sy = (gy == 0 || gy == HD - 1) ? 1.0f : 0.5f;
    const float sz = (gz == 0 || gz == HD - 1) ? 1.0f : 0.5f;

    float X00, X01, X02, X10, X11, X12, X20, X21, X22;
    {
        X00 = (SD(0, lx + 2, ly + 1, lz + 1) - SD(0, lx, ly + 1, lz + 1)) * sx + 1.0f;
        X01 = (SD(0, lx + 1, ly + 2, lz + 1) - SD(0, lx + 1, ly, lz + 1)) * sy;
        X02 = (SD(0, lx + 1, ly + 1, lz + 2) - SD(0, lx + 1, ly + 1, lz)) * sz;
        X10 = (SD(1, lx + 2, ly + 1, lz + 1) - SD(1, lx, ly + 1, lz + 1)) * sx;
        X11 = (SD(1, lx + 1, ly + 2, lz + 1) - SD(1, lx + 1, ly, lz + 1)) * sy + 1.0f;
        X12 = (SD(1, lx + 1, ly + 1, lz + 2) - SD(1, lx + 1, ly + 1, lz)) * sz;
        X20 = (SD(2, lx + 2, ly + 1, lz + 1) - SD(2, lx, ly + 1, lz + 1)) * sx;
        X21 = (SD(2, lx + 1, ly + 2, lz + 1) - SD(2, lx + 1, ly, lz + 1)) * sy;
        X22 = (SD(2, lx + 1, ly + 1, lz + 2) - SD(2, lx + 1, ly + 1, lz)) * sz + 1.0f;
    }
#undef SD

    // ---------------------------------------------------------------
    // Trilinear warp with border padding of the 6 DTI channels.
    // z-corner pairs fetched as one dword-aligned b64 load; at the
    // z==63 border the z1 weight is exactly 0, so load at z=62 and
    // select the high lane for the z0 term.
    // ---------------------------------------------------------------
    float cx = fminf(fmaxf((float)gx + u0, 0.0f), (float)(HD - 1));
    float cy = fminf(fmaxf((float)gy + u1, 0.0f), (float)(HD - 1));
    float cz = fminf(fmaxf((float)gz + u2, 0.0f), (float)(HD - 1));
    float xf = floorf(cx), yf = floorf(cy), zf = floorf(cz);
    float fx = cx - xf, fy = cy - yf, fz = cz - zf;
    int x0i = (int)xf, y0i = (int)yf, z0i = (int)zf;
    int x1i = min(x0i + 1, HD - 1);
    int y1i = min(y0i + 1, HD - 1);

    const int   xs[2] = { x0i, x1i };   const float wx[2] = { 1.0f - fx, fx };
    const int   ys[2] = { y0i, y1i };   const float wy[2] = { 1.0f - fy, fy };
    const float wz0 = 1.0f - fz, wz1 = fz;

    const int  zb  = min(z0i, HD - 2);
    const bool zhi = (z0i == HD - 1);   // then fz == 0, z1 term weightless

    const float* dtib = dti + (size_t)b * 6 * S3;
    float m[6] = { 0.f, 0.f, 0.f, 0.f, 0.f, 0.f };
#pragma unroll
    for (int ii = 0; ii < 2; ++ii) {
#pragma unroll
        for (int jj = 0; jj < 2; ++jj) {
            const float wxy = wx[ii] * wy[jj];
            const float w0  = wxy * wz0;
            const float w1  = wxy * wz1;
            const float* p  = dtib + ((xs[ii] * HD + ys[jj]) * HD + zb);
#pragma unroll
            for (int c = 0; c < 6; ++c) {
                uf2 v = *(const uf2*)(p + (size_t)c * S3);
                float v0 = zhi ? v.y : v.x;   // value at z0
                float v1 = v.y;               // value at z1 (exact, or weight 0)
                m[c] = fmaf(v0, w0, m[c]);
                m[c] = fmaf(v1, w1, m[c]);
            }
        }
    }

    // ---------------------------------------------------------------
    // R = orthogonal polar factor of J  (== U Vh of the SVD)
    // Determinant-scaled Newton: X <- 0.5*(g*X + cof(X)/(g*det))
    // ---------------------------------------------------------------
#pragma unroll
    for (int it = 0; it < 12; ++it) {
        float C00 = X11 * X22 - X12 * X21;
        float C01 = X12 * X20 - X10 * X22;
        float C02 = X10 * X21 - X11 * X20;
        float C10 = X02 * X21 - X01 * X22;
        float C11 = X00 * X22 - X02 * X20;
        float C12 = X01 * X20 - X00 * X21;
        float C20 = X01 * X12 - X02 * X11;
        float C21 = X02 * X10 - X00 * X12;
        float C22 = X00 * X11 - X01 * X10;
        float det  = X00 * C00 + X01 * C01 + X02 * C02;
        float adet = fmaxf(fabsf(det), 1e-20f);
        float dsgn = (det < 0.0f) ? -adet : adet;
        float g    = exp2f(-0.3333333333f * log2f(adet));  // |det|^(-1/3)
        float s    = 0.5f * g;
        float t    = 0.5f / (g * dsgn);
        X00 = s * X00 + t * C00;  X01 = s * X01 + t * C01;  X02 = s * X02 + t * C02;
        X10 = s * X10 + t * C10;  X11 = s * X11 + t * C11;  X12 = s * X12 + t * C12;
        X20 = s * X20 + t * C20;  X21 = s * X21 + t * C21;  X22 = s * X22 + t * C22;
    }

    // ---------------------------------------------------------------
    // A = R^T * M * R  with symmetric M from warped channels
    // ---------------------------------------------------------------
    const float M00 = m[0], M01 = m[1], M11 = m[2], M02 = m[3], M12 = m[4], M22 = m[5];

    const float T00 = X00 * M00 + X10 * M01 + X20 * M02;
    const float T01 = X00 * M01 + X10 * M11 + X20 * M12;
    const float T02 = X00 * M02 + X10 * M12 + X20 * M22;
    const float T10 = X01 * M00 + X11 * M01 + X21 * M02;
    const float T11 = X01 * M01 + X11 * M11 + X21 * M12;
    const float T12 = X01 * M02 + X11 * M12 + X21 * M22;
    const float T20 = X02 * M00 + X12 * M01 + X22 * M02;
    const float T21 = X02 * M01 + X12 * M11 + X22 * M12;
    const float T22 = X02 * M02 + X12 * M12 + X22 * M22;

    const float A00 = T00 * X00 + T01 * X10 + T02 * X20;
    const float A10 = T10 * X00 + T11 * X10 + T12 * X20;
    const float A11 = T10 * X01 + T11 * X11 + T12 * X21;
    const float A20 = T20 * X00 + T21 * X10 + T22 * X20;
    const float A21 = T20 * X01 + T21 * X11 + T22 * X21;
    const float A22 = T20 * X02 + T21 * X12 + T22 * X22;

    // Lower-tri pack: channels = A00, A10, A11, A20, A21, A22
    float* ob = out + (size_t)b * 6 * S3 + (((size_t)gx * HD + gy) * HD + gz);
    ob[0 * S3] = A00;
    ob[1 * S3] = A10;
    ob[2 * S3] = A11;
    ob[3 * S3] = A20;
    ob[4 * S3] = A21;
    ob[5 * S3] = A22;
}

extern "C" void kernel_launch(void* const* d_in, const int* in_sizes, int n_in,
                              void* d_out, int out_size, void* d_ws, size_t ws_size,
                              hipStream_t stream) {
    (void)n_in; (void)d_ws; (void)ws_size; (void)out_size;
    const float* dti = (const float*)d_in[0];   // (B,6,64,64,64) f32
    const float* ddf = (const float*)d_in[1];   // (B,3,64,64,64) f32
    float* out = (float*)d_out;                 // (B,6,64,64,64) f32
    const int B = in_sizes[0] / (6 * S3);       // = 2
    // 1024 tiles (16 x 8 x 8) per batch, 256 threads per tile
    warp_dti_polar_kernel<<<B * 1024, 256, 0, stream>>>(dti, ddf, out);
}